// HyperbolicGraphConvolution_171798692373
// MI455X (gfx1250) — compile-verified
//
#include <hip/hip_runtime.h>
#include <hip/hip_bf16.h>

#define DFEAT 128
#define WPB   8           // waves per 256-thread block (wave32)
#define PIPE  4           // outstanding async gathers per wave

__device__ __forceinline__ float wave_sum(float v) {
#pragma unroll
    for (int off = 16; off > 0; off >>= 1)
        v += __shfl_xor(v, off, 32);
    return v;
}

// -------- h = logmap0(x):  h[0]=0, h[1:] = acosh(max(x0,1+eps)) * y/||y|| -----
__global__ void logmap_kernel(const float* __restrict__ x,
                              float* __restrict__ h, int n) {
    const int w    = threadIdx.x >> 5;
    const int lane = threadIdx.x & 31;
    const int row  = blockIdx.x * WPB + w;
    if (row >= n) return;

    const float4 v = ((const float4*)(x + (size_t)row * DFEAT))[lane];
    const float x0 = __shfl(v.x, 0, 32);               // x[row][0]

    float sq = v.y * v.y + v.z * v.z + v.w * v.w;
    if (lane != 0) sq += v.x * v.x;                    // exclude x[row][0]
    const float sum   = wave_sum(sq);
    const float ynorm = fmaxf(sqrtf(sum), 1e-15f);
    const float theta = fmaxf(x0, 1.0f + 1e-5f);
    const float scale = acoshf(theta) / ynorm;         // sqrtK = 1 (C = 1)

    float4 o;
    o.x = (lane == 0) ? 0.0f : v.x * scale;
    o.y = v.y * scale;  o.z = v.z * scale;  o.w = v.w * scale;
    ((float4*)(h + (size_t)row * DFEAT))[lane] = o;
}

// -------- zero fill for int buffers ------------------------------------------
__global__ void zero_int_kernel(int* __restrict__ p, int n) {
    for (int i = blockIdx.x * blockDim.x + threadIdx.x; i < n;
         i += gridDim.x * blockDim.x)
        p[i] = 0;
}

// -------- CSR build step 1: per-row degree histogram --------------------------
__global__ void hist_kernel(const int* __restrict__ rows,
                            int* __restrict__ counts, int E) {
    const int i = blockIdx.x * blockDim.x + threadIdx.x;
    if (i < E) atomicAdd(&counts[rows[i]], 1);
}

// -------- CSR build step 2: exclusive scan of counts -> rowStart --------------
// Single block of 1024 threads (32 wave32s); n = 50000 -> 49 items/thread.
__global__ void scan_kernel(const int* __restrict__ counts,
                            int* __restrict__ rowStart, int n) {
    __shared__ int wsum[32];
    const int t     = threadIdx.x;
    const int lane  = t & 31;
    const int wid   = t >> 5;
    const int items = (n + 1023) / 1024;
    const int base  = t * items;

    int s = 0;
    for (int k = 0; k < items; k++) {
        int idx = base + k;
        if (idx < n) s += counts[idx];
    }
    // block-wide exclusive scan of per-thread sums
    int v = s;
#pragma unroll
    for (int off = 1; off < 32; off <<= 1) {
        int y = __shfl_up(v, off, 32);
        if (lane >= off) v += y;
    }
    if (lane == 31) wsum[wid] = v;
    __syncthreads();
    if (wid == 0) {
        int wv = wsum[lane];
#pragma unroll
        for (int off = 1; off < 32; off <<= 1) {
            int y = __shfl_up(wv, off, 32);
            if (lane >= off) wv += y;
        }
        wsum[lane] = wv;
    }
    __syncthreads();
    const int waveOff   = (wid > 0) ? wsum[wid - 1] : 0;
    const int exclusive = waveOff + v - s;

    int run = exclusive;
    for (int k = 0; k < items; k++) {
        int idx = base + k;
        if (idx < n) { rowStart[idx] = run; run += counts[idx]; }
    }
    if (t == 1023) rowStart[n] = run;          // == E
}

// -------- CSR build step 3: scatter edges into row-sorted order ---------------
__global__ void scatter_kernel(const int* __restrict__ rows,
                               const int* __restrict__ cols,
                               const float* __restrict__ vals,
                               const int* __restrict__ rowStart,
                               int* __restrict__ fill,
                               int* __restrict__ scol,
                               float* __restrict__ sval, int E) {
    const int i = blockIdx.x * blockDim.x + threadIdx.x;
    if (i < E) {
        const int r   = rows[i];
        const int pos = rowStart[r] + atomicAdd(&fill[r], 1);
        scol[pos] = cols[i];
        sval[pos] = vals[i];
    }
}

// -------- pull-based SpMM: hout[r] = sum_e val[e] * hin[col[e]] ---------------
// One wave32 per row, no atomics. Row gathers (512B each) stream through a
// 4-deep LDS ring via CDNA5 async global->LDS DMA: 4 outstanding per wave,
// steady-state s_wait_asynccnt 3 retires the oldest (in-order completion).
__global__ void spmm_pull_kernel(const int* __restrict__ rowStart,
                                 const int* __restrict__ scol,
                                 const float* __restrict__ sval,
                                 const float* __restrict__ hin,
                                 float* __restrict__ hout, int n) {
    __shared__ __align__(16) float lds[WPB * PIPE * DFEAT];   // 8 x 4 x 512B = 16KB

    const int w    = threadIdx.x >> 5;
    const int lane = threadIdx.x & 31;
    const int row  = blockIdx.x * WPB + w;
    if (row >= n) return;

    const int beg = rowStart[row];
    const int end = rowStart[row + 1];
    const int cnt = end - beg;

    float* base = &lds[(w * PIPE) * DFEAT] + lane * 4;        // slot stride = 512B
    const unsigned base_off = (unsigned)(unsigned long long)base;

    float4 acc = make_float4(0.f, 0.f, 0.f, 0.f);

    // prime: up to PIPE async gathers in flight
    const int npre = (cnt < PIPE) ? cnt : PIPE;
    for (int k = 0; k < npre; ++k) {
        const float*   src = hin + (size_t)scol[beg + k] * DFEAT + lane * 4;
        const unsigned o   = base_off + (unsigned)(k * DFEAT * 4);
        asm volatile("global_load_async_to_lds_b128 %0, %1, off"
                     :: "v"(o), "v"(src) : "memory");
    }

    int i = beg;
    const int steady = end - PIPE;
    for (; i < steady; ++i) {                 // 4 outstanding; oldest has landed
        asm volatile("s_wait_asynccnt 3" ::: "memory");
        const int    slot = (i - beg) & (PIPE - 1);
        const float  v    = sval[i];
        const float4 hv   = *(const float4*)(base + slot * DFEAT);
        acc.x += v * hv.x;  acc.y += v * hv.y;
        acc.z += v * hv.z;  acc.w += v * hv.w;
        // refill the slot just consumed with edge i+PIPE
        const float*   src = hin + (size_t)scol[i + PIPE] * DFEAT + lane * 4;
        const unsigned o   = base_off + (unsigned)(slot * DFEAT * 4);
        asm volatile("s_wait_dscnt 0\n\t"     // ds_load of this slot fully drained
                     "global_load_async_to_lds_b128 %0, %1, off"
                     :: "v"(o), "v"(src) : "memory");
    }
    for (; i < end; ++i) {                    // tail: wait-all (<= PIPE edges)
        asm volatile("s_wait_asynccnt 0" ::: "memory");
        const int    slot = (i - beg) & (PIPE - 1);
        const float  v    = sval[i];
        const float4 hv   = *(const float4*)(base + slot * DFEAT);
        acc.x += v * hv.x;  acc.y += v * hv.y;
        acc.z += v * hv.z;  acc.w += v * hv.w;
    }
    ((float4*)(hout + (size_t)row * DFEAT))[lane] = acc;
}

// -------- out = proj(expmap0(h)) ---------------------------------------------
__global__ void expproj_kernel(const float* __restrict__ h,
                               float* __restrict__ out, int n) {
    const int w    = threadIdx.x >> 5;
    const int lane = threadIdx.x & 31;
    const int row  = blockIdx.x * WPB + w;
    if (row >= n) return;

    const float4 v = ((const float4*)(h + (size_t)row * DFEAT))[lane];

    float sq = v.y * v.y + v.z * v.z + v.w * v.w;
    if (lane != 0) sq += v.x * v.x;                    // v = h[1:]
    const float vn2   = wave_sum(sq);
    const float vnorm = fmaxf(sqrtf(vn2), 1e-15f);
    const float s     = sinhf(vnorm) / vnorm;          // sqrtK = 1

    float4 t;
    t.x = v.x * s;  t.y = v.y * s;  t.z = v.z * s;  t.w = v.w * s;

    float tsq = t.y * t.y + t.z * t.z + t.w * t.w;
    if (lane != 0) tsq += t.x * t.x;
    const float ysq   = wave_sum(tsq);
    const float first = sqrtf(fmaxf(1.0f + ysq, 1e-5f));

    float4 o;
    o.x = (lane == 0) ? first : t.x;
    o.y = t.y;  o.z = t.z;  o.w = t.w;
    ((float4*)(out + (size_t)row * DFEAT))[lane] = o;
}

extern "C" void kernel_launch(void* const* d_in, const int* in_sizes, int n_in,
                              void* d_out, int out_size, void* d_ws, size_t ws_size,
                              hipStream_t stream) {
    const float* x    = (const float*)d_in[0];
    const int*   rows = (const int*)  d_in[1];
    const int*   cols = (const int*)  d_in[2];
    const float* vals = (const float*)d_in[3];
    float*       out  = (float*)d_out;

    const int N = in_sizes[0] / DFEAT;
    const int E = in_sizes[1];

    // ---- workspace layout (all 4B-aligned; hws 16B-aligned at base) ----
    float* hws      = (float*)d_ws;                     // N*128 floats (25.6 MB)
    int*   counts   = (int*)(hws + (size_t)N * DFEAT);  // N
    int*   fill     = counts + N;                       // N
    int*   rowStart = fill + N;                         // N+1
    int*   scol     = rowStart + N + 1;                 // E
    float* sval     = (float*)(scol + E);               // E

    const int rowBlocks  = (N + WPB - 1) / WPB;
    const int edgeBlocks = (E + 255) / 256;

    // ---- build CSR once (reused by all 3 layers) ----
    zero_int_kernel<<<512, 256, 0, stream>>>(counts, 2 * N);   // counts+fill
    hist_kernel<<<edgeBlocks, 256, 0, stream>>>(rows, counts, E);
    scan_kernel<<<1, 1024, 0, stream>>>(counts, rowStart, N);
    scatter_kernel<<<edgeBlocks, 256, 0, stream>>>(rows, cols, vals, rowStart,
                                                   fill, scol, sval, E);

    // ---- h0 = logmap0(x) -> d_out (ping buffer) ----
    logmap_kernel<<<rowBlocks, 256, 0, stream>>>(x, out, N);

    // ---- 3 pull-SpMM layers, ping-pong d_out <-> hws ----
    spmm_pull_kernel<<<rowBlocks, 256, 0, stream>>>(rowStart, scol, sval, out, hws, N);
    spmm_pull_kernel<<<rowBlocks, 256, 0, stream>>>(rowStart, scol, sval, hws, out, N);
    spmm_pull_kernel<<<rowBlocks, 256, 0, stream>>>(rowStart, scol, sval, out, hws, N);

    // ---- out = proj(expmap0(h3)) ----
    expproj_kernel<<<rowBlocks, 256, 0, stream>>>(hws, out, N);
}